// LSTMModel_1700807050126
// MI455X (gfx1250) — compile-verified
//
#include <hip/hip_runtime.h>
#include <hip/hip_bf16.h>

#define HIDDEN  128
#define T_STEPS 512
#define BATCH   512

typedef __attribute__((ext_vector_type(2))) float v2f;
typedef __attribute__((ext_vector_type(8))) float v8f;

// gfx1250 native transcendental tanh (CDNA5-only TRANS op).
__device__ __forceinline__ float fast_tanh(float x) {
#if __has_builtin(__builtin_amdgcn_tanhf)
    return __builtin_amdgcn_tanhf(x);
#elif __has_builtin(__builtin_amdgcn_tanh_f32)
    return __builtin_amdgcn_tanh_f32(x);
#else
    float y;
    asm("v_tanh_f32 %0, %1\n\tv_nop" : "=v"(y) : "v"(x));
    return y;
#endif
}

// sigmoid(x) = 0.5*tanh(0.5x) + 0.5  -> single TRANS op
__device__ __forceinline__ float fast_sigmoid(float x) {
    return __builtin_fmaf(0.5f, fast_tanh(0.5f * x), 0.5f);
}

__global__ __launch_bounds__(256) void lstm_autonomous_kernel(
    const float* __restrict__ x,     // [T, B, 1] (only t=0 slice used)
    const float* __restrict__ W_ih,  // [4H, 1]
    const float* __restrict__ b_ih,  // [4H]
    const float* __restrict__ b_hh,  // [4H]
    const float* __restrict__ fc_w,  // [1, H]
    const float* __restrict__ fc_b,  // [1]
    float* __restrict__ out)         // [T, B]
{
    const int lane = threadIdx.x & 31;
    const int wave = threadIdx.x >> 5;
    const int b    = blockIdx.x * 8 + wave;   // one wave32 per batch chain

    // Per-lane channels k = lane + 32*j. Gate rows: i=[0,H), f unused, g=[2H,3H), o=[3H,4H).
    float Wi[4], Wg[4], Wo[4], Bi[4], Bg[4], Bo[4], Fw[4];
#pragma unroll
    for (int j = 0; j < 4; ++j) {
        const int k = lane + 32 * j;
        Wi[j] = W_ih[k];
        Wg[j] = W_ih[2 * HIDDEN + k];
        Wo[j] = W_ih[3 * HIDDEN + k];
        Bi[j] = b_ih[k]              + b_hh[k];
        Bg[j] = b_ih[2 * HIDDEN + k] + b_hh[2 * HIDDEN + k];
        Bo[j] = b_ih[3 * HIDDEN + k] + b_hh[3 * HIDDEN + k];
        Fw[j] = fc_w[k];
    }
    const float fb = fc_b[0];

    float p = x[b];                       // x[0, b, 0]
    const v2f bones = {1.0f, 1.0f};       // B(4x16) = all ones

    for (int t0 = 0; t0 < T_STEPS; t0 += 32) {
        float buf = 0.0f;                 // lane l keeps step (t0 + l)

        for (int tt = 0; tt < 32; ++tt) {
            // 4 channels per lane; keep two independent partial pairs so the
            // WMMA K-sum absorbs the final reduction level.
            float r[4];
#pragma unroll
            for (int j = 0; j < 4; ++j) {
                const float iv = fast_sigmoid(__builtin_fmaf(p, Wi[j], Bi[j]));
                const float gv = fast_tanh  (__builtin_fmaf(p, Wg[j], Bg[j]));
                const float ov = fast_sigmoid(__builtin_fmaf(p, Wo[j], Bo[j]));
                const float c  = iv * gv;
                const float h  = ov * fast_tanh(c);
                r[j] = fmaxf(h, 0.0f);
            }

            // WMMA-1: A K-slots carry the two pair-sums directly
            //   lanes 0-15 : a1 = {q0,q1} -> A[m,0], A[m,1]
            //   lanes16-31 : a1 = {q0,q1} -> A[m,2], A[m,3]
            // B = ones => d[r] = S_r = full per-row channel sums.
            v2f a1;
            a1.x = __builtin_fmaf(r[0], Fw[0], r[1] * Fw[1]);
            a1.y = __builtin_fmaf(r[2], Fw[2], r[3] * Fw[3]);
            v8f cz = {};
            v8f d = __builtin_amdgcn_wmma_f32_16x16x4_f32(
                false, a1, false, bones, (short)0, cz, false, false);

            // WMMA-2: fold the 16 row sums, broadcast wave total to ALL lanes.
            v2f a2;
            a2.x = (d[0] + d[1]) + (d[2] + d[3]);
            a2.y = (d[4] + d[5]) + (d[6] + d[7]);
            v8f d2 = __builtin_amdgcn_wmma_f32_16x16x4_f32(
                false, a2, false, bones, (short)0, cz, false, false);

            p = d2[0] + fb;               // uniform in every lane
            buf = (lane == tt) ? p : buf; // branchless capture
        }

        out[(t0 + lane) * BATCH + b] = buf;  // one store per 32 steps per lane
    }
}

extern "C" void kernel_launch(void* const* d_in, const int* in_sizes, int n_in,
                              void* d_out, int out_size, void* d_ws, size_t ws_size,
                              hipStream_t stream) {
    (void)in_sizes; (void)n_in; (void)d_ws; (void)ws_size; (void)out_size;
    const float* x    = (const float*)d_in[0];
    const float* W_ih = (const float*)d_in[1];
    // d_in[2] = W_hh: mathematically dead (h0 == 0 every step)
    const float* b_ih = (const float*)d_in[3];
    const float* b_hh = (const float*)d_in[4];
    const float* fc_w = (const float*)d_in[5];
    const float* fc_b = (const float*)d_in[6];
    float* out = (float*)d_out;

    dim3 grid(BATCH / 8);   // 64 blocks
    dim3 block(256);        // 8 waves/block, one wave32 per batch chain
    lstm_autonomous_kernel<<<grid, block, 0, stream>>>(x, W_ih, b_ih, b_hh, fc_w, fc_b, out);
}